// MemoryEfficientPacoRefinementModule_2216203125203
// MI455X (gfx1250) — compile-verified
//
#include <hip/hip_runtime.h>
#include <hip/hip_bf16.h>
#include <math.h>

// ---------------------------------------------------------------------------
// MI455X (gfx1250) implementation.
// f32 WMMA (V_WMMA_F32_16X16X4_F32) for all GEMMs: workload is ~10 GFLOP /
// ~50 MB => latency bound, so keep reference f32 precision on matrix pipes.
// ---------------------------------------------------------------------------

typedef __attribute__((ext_vector_type(2))) float v2f;
typedef __attribute__((ext_vector_type(8))) float v8f;

#define KNN_K   16
#define THRESH  0.1f

// ---------------------------------------------------------------------------
// Wave-level f32 WMMA K-loop: computes one 16x16 C tile accumulating
//   C += A(16 x F, LDS row-major) * W(rows x ldw, global)[:, colBase:colBase+16]
// A fragment layout (16x4 f32): lane m=l%16 holds row m; VGPR0 K=k0+(l<16?0:2),
// VGPR1 K=+1. B fragment (4x16): lane holds col m; VGPR0 row K=k0+(l<16?0:2).
// ---------------------------------------------------------------------------
__device__ inline v8f wmma_col(const float* __restrict__ sA, int F,
                               const float* __restrict__ W, int Wrows, int ldw,
                               int colBase, int lane, v8f acc) {
  const int m  = lane & 15;
  const int kh = (lane >> 4) << 1;   // 0 or 2
  for (int k0 = 0; k0 < F; k0 += 4) {
    const int r0 = k0 + kh;
    v2f a, bb;
    a.x = sA[m * F + r0];
    a.y = sA[m * F + r0 + 1];
    bb.x = (r0     < Wrows) ? W[(size_t)(r0    ) * ldw + colBase + m] : 0.0f;
    bb.y = (r0 + 1 < Wrows) ? W[(size_t)(r0 + 1) * ldw + colBase + m] : 0.0f;
    acc = __builtin_amdgcn_wmma_f32_16x16x4_f32(false, a, false, bb,
                                                (short)0, acc, false, false);
  }
  return acc;
}

__device__ inline v8f zero8() {
  v8f z = {0.f, 0.f, 0.f, 0.f, 0.f, 0.f, 0.f, 0.f};
  return z;
}

// ---------------------------------------------------------------------------
// 1) Brute-force kNN, K=16. One thread per query point, LDS tiles of 256 pts.
// ---------------------------------------------------------------------------
__global__ __launch_bounds__(256) void knn_kernel(const float* __restrict__ pts,
                                                  int* __restrict__ knn, int N_) {
  __shared__ float sx[256], sy[256], sz[256];
  const int b = blockIdx.y;
  const int i = blockIdx.x * 256 + threadIdx.x;
  const float* p = pts + ((size_t)b * N_ + i) * 3;
  const float px = p[0], py = p[1], pz = p[2];

  float bd[KNN_K];
  int   bi[KNN_K];
#pragma unroll
  for (int k = 0; k < KNN_K; ++k) { bd[k] = 3.0e38f; bi[k] = 0; }

  for (int t = 0; t < N_; t += 256) {
    const float* q = pts + ((size_t)b * N_ + t + threadIdx.x) * 3;
    sx[threadIdx.x] = q[0]; sy[threadIdx.x] = q[1]; sz[threadIdx.x] = q[2];
    __syncthreads();
    for (int jj = 0; jj < 256; ++jj) {
      const int j2 = t + jj;
      const float dx = px - sx[jj], dy = py - sy[jj], dz = pz - sz[jj];
      const float d = dx * dx + dy * dy + dz * dz;
      if (j2 != i && d < bd[KNN_K - 1]) {
        int pp = KNN_K - 1;
        while (pp > 0 && bd[pp - 1] > d) {
          bd[pp] = bd[pp - 1]; bi[pp] = bi[pp - 1]; --pp;
        }
        bd[pp] = d; bi[pp] = j2;
      }
    }
    __syncthreads();
  }
  int* o = knn + ((size_t)b * N_ + i) * KNN_K;
#pragma unroll
  for (int k = 0; k < KNN_K; ++k) o[k] = bi[k];
}

// ---------------------------------------------------------------------------
// 2) EdgeConv: one wave per point (its 16 edges are exactly a 16-row WMMA
// tile). f = [xi, xj-xi] (2*CIN, padded to FPAD), h1 = relu(f@w1+b1),
// h2 = h1@w2+b2, out = max over the 16 edges (rows).
// ---------------------------------------------------------------------------
template <int CIN, int FPAD>
__global__ __launch_bounds__(128) void edgeconv_kernel(
    const float* __restrict__ x, const int* __restrict__ idx,
    const float* __restrict__ w1, const float* __restrict__ b1,
    const float* __restrict__ w2, const float* __restrict__ b2,
    float* __restrict__ out, int N_) {
  constexpr int WAVES = 4;
  __shared__ float sF[WAVES][16 * FPAD];
  __shared__ float sH[WAVES][16 * 64];

  const int wave  = threadIdx.x >> 5;
  const int lane  = threadIdx.x & 31;
  const int point = blockIdx.x * WAVES + wave;          // global over B*N
  const int b     = point / N_;
  const int* nbr  = idx + (size_t)point * KNN_K;
  const float* xp = x + (size_t)point * CIN;

  // Stage 16 x FPAD edge-feature tile (zero-padded past 2*CIN).
  for (int e = lane; e < 16 * FPAD; e += 32) {
    const int k = e / FPAD, c = e - k * FPAD;
    float v = 0.0f;
    if (c < CIN) {
      v = xp[c];
    } else if (c < 2 * CIN) {
      const int cc = c - CIN;
      const int j  = nbr[k];
      v = x[((size_t)b * N_ + j) * CIN + cc] - xp[cc];
    }
    sF[wave][e] = v;
  }
  __syncthreads();

  const int rb = (lane >> 4) << 3;   // row base 0 or 8 in C layout
  const int cm = lane & 15;

  // GEMM1: (16 x FPAD) * (2*CIN x 64) -> h1, bias + relu into LDS.
  {
    v8f acc[4];
#pragma unroll
    for (int n = 0; n < 4; ++n) acc[n] = zero8();
#pragma unroll
    for (int n = 0; n < 4; ++n)
      acc[n] = wmma_col(sF[wave], FPAD, w1, 2 * CIN, 64, n * 16, lane, acc[n]);
#pragma unroll
    for (int n = 0; n < 4; ++n) {
      const int col = n * 16 + cm;
      const float bias = b1[col];
#pragma unroll
      for (int v = 0; v < 8; ++v)
        sH[wave][(v + rb) * 64 + col] = fmaxf(acc[n][v] + bias, 0.0f);
    }
  }
  __syncthreads();

  // GEMM2: (16 x 64) * (64 x 64) -> h2, bias + max over 16 rows -> out[64].
  {
    v8f acc[4];
#pragma unroll
    for (int n = 0; n < 4; ++n) acc[n] = zero8();
#pragma unroll
    for (int n = 0; n < 4; ++n)
      acc[n] = wmma_col(sH[wave], 64, w2, 64, 64, n * 16, lane, acc[n]);
#pragma unroll
    for (int n = 0; n < 4; ++n) {
      const int col = n * 16 + cm;
      const float bias = b2[col];
      float mv = -3.0e38f;
#pragma unroll
      for (int v = 0; v < 8; ++v) mv = fmaxf(mv, acc[n][v] + bias);
      mv = fmaxf(mv, __shfl_xor(mv, 16, 32));   // rows 0-7 vs 8-15 halves
      if (lane < 16) out[(size_t)point * 64 + col] = mv;
    }
  }
}

// ---------------------------------------------------------------------------
// 3) MLP head: feat = concat(f1,f2,f3) (192) -> 256 relu -> 3.
// One wave per 16-point tile; GEMM1 via WMMA (16 col-tiles), GEMM2 scalar.
// Writes residual and pts_res = points + residual.
// ---------------------------------------------------------------------------
__global__ __launch_bounds__(64) void mlp_kernel(
    const float* __restrict__ f1, const float* __restrict__ f2,
    const float* __restrict__ f3,
    const float* __restrict__ w1, const float* __restrict__ b1,
    const float* __restrict__ w2, const float* __restrict__ b2,
    const float* __restrict__ points,
    float* __restrict__ residual_out, float* __restrict__ pts_res) {
  constexpr int WAVES = 2;
  __shared__ float sFe[WAVES][16 * 192];
  __shared__ float sH[WAVES][16 * 256];

  const int wave = threadIdx.x >> 5;
  const int lane = threadIdx.x & 31;
  const int base = (blockIdx.x * WAVES + wave) * 16;   // first point of tile

  for (int e = lane; e < 16 * 192; e += 32) {
    const int r = e / 192, c = e - r * 192;
    const size_t pt = (size_t)(base + r);
    float v;
    if (c < 64)        v = f1[pt * 64 + c];
    else if (c < 128)  v = f2[pt * 64 + (c - 64)];
    else               v = f3[pt * 64 + (c - 128)];
    sFe[wave][e] = v;
  }
  __syncthreads();

  const int rb = (lane >> 4) << 3;
  const int cm = lane & 15;

  for (int nt = 0; nt < 16; ++nt) {
    v8f acc = zero8();
    acc = wmma_col(sFe[wave], 192, w1, 192, 256, nt * 16, lane, acc);
    const int col = nt * 16 + cm;
    const float bias = b1[col];
#pragma unroll
    for (int v = 0; v < 8; ++v)
      sH[wave][(v + rb) * 256 + col] = fmaxf(acc[v] + bias, 0.0f);
  }
  __syncthreads();

  // 256 -> 3 scalar tail.
#pragma unroll
  for (int p = 0; p < 2; ++p) {
    const int m = lane & 15;
    const int dcol = (lane >> 4) + 2 * p;
    if (dcol < 3) {
      float s = b2[dcol];
      const float* hr = &sH[wave][m * 256];
      for (int c = 0; c < 256; ++c) s = fmaf(hr[c], w2[c * 3 + dcol], s);
      const size_t pt = (size_t)(base + m);
      residual_out[pt * 3 + dcol] = s;
      pts_res[pt * 3 + dcol] = points[pt * 3 + dcol] + s;
    }
  }
}

// ---------------------------------------------------------------------------
// 4) Per-plane stats: mask/centroid/cov over pts_res (independent of scan
// state in the reference), then 3x3 Jacobi eigen and rn = row-2 of V
// (replicating the reference's vh[:,2] indexing on the symmetric cov).
// One block per (plane, batch).
// stats layout per (b,m), stride 12: {ux,uy,uz,d, rnx,rny,rnz,rd, valid}
// ---------------------------------------------------------------------------
__device__ inline void jacobi_rot(float A[3][3], float Q[3][3], int p, int q) {
  const float apq = A[p][q];
  if (fabsf(apq) < 1e-20f) return;
  const float theta = (A[q][q] - A[p][p]) / (2.0f * apq);
  float t = 1.0f / (fabsf(theta) + sqrtf(theta * theta + 1.0f));
  if (theta < 0.0f) t = -t;
  const float c = 1.0f / sqrtf(t * t + 1.0f);
  const float s = t * c;
  for (int k = 0; k < 3; ++k) {
    const float akp = A[k][p], akq = A[k][q];
    A[k][p] = c * akp - s * akq;
    A[k][q] = s * akp + c * akq;
  }
  for (int k = 0; k < 3; ++k) {
    const float apk = A[p][k], aqk = A[q][k];
    A[p][k] = c * apk - s * aqk;
    A[q][k] = s * apk + c * aqk;
  }
  for (int k = 0; k < 3; ++k) {
    const float qkp = Q[k][p], qkq = Q[k][q];
    Q[k][p] = c * qkp - s * qkq;
    Q[k][q] = s * qkp + c * qkq;
  }
}

__global__ __launch_bounds__(256) void plane_stats_kernel(
    const float* __restrict__ planes, const float* __restrict__ pr,
    float* __restrict__ stats, int N_, int M_) {
  __shared__ float red[6][256];
  __shared__ float sc[4];
  const int m = blockIdx.x, b = blockIdx.y;
  const int tid = threadIdx.x;

  const float* pl = planes + ((size_t)b * M_ + m) * 4;
  const float nx = pl[0], ny = pl[1], nz = pl[2], pd = pl[3];
  const float nrm = sqrtf(nx * nx + ny * ny + nz * nz);
  const float inv = 1.0f / fmaxf(nrm, 1e-12f);
  const float ux = nx * inv, uy = ny * inv, uz = nz * inv;
  const float* P = pr + (size_t)b * N_ * 3;

  // Pass 1: count + centroid sums.
  float cx = 0.f, cy = 0.f, cz = 0.f, cn = 0.f;
  for (int i = tid; i < N_; i += 256) {
    const float x = P[i * 3], y = P[i * 3 + 1], z = P[i * 3 + 2];
    const float dist = fabsf(x * ux + y * uy + z * uz + pd);
    if (dist < THRESH) { cx += x; cy += y; cz += z; cn += 1.f; }
  }
  red[0][tid] = cx; red[1][tid] = cy; red[2][tid] = cz; red[3][tid] = cn;
  __syncthreads();
  for (int s = 128; s >= 1; s >>= 1) {
    if (tid < s)
      for (int q = 0; q < 4; ++q) red[q][tid] += red[q][tid + s];
    __syncthreads();
  }
  if (tid == 0) {
    const float cnt = red[3][0];
    const float div = fmaxf(cnt, 1.0f);
    sc[0] = red[0][0] / div; sc[1] = red[1][0] / div; sc[2] = red[2][0] / div;
    sc[3] = cnt;
  }
  __syncthreads();
  const float mx = sc[0], my = sc[1], mz = sc[2], cnt = sc[3];

  // Pass 2: masked, centered covariance (6 unique terms).
  float c00 = 0.f, c01 = 0.f, c02 = 0.f, c11 = 0.f, c12 = 0.f, c22 = 0.f;
  for (int i = tid; i < N_; i += 256) {
    const float x = P[i * 3], y = P[i * 3 + 1], z = P[i * 3 + 2];
    const float dist = fabsf(x * ux + y * uy + z * uz + pd);
    if (dist < THRESH) {
      const float ax = x - mx, ay = y - my, az = z - mz;
      c00 += ax * ax; c01 += ax * ay; c02 += ax * az;
      c11 += ay * ay; c12 += ay * az; c22 += az * az;
    }
  }
  red[0][tid] = c00; red[1][tid] = c01; red[2][tid] = c02;
  red[3][tid] = c11; red[4][tid] = c12; red[5][tid] = c22;
  __syncthreads();
  for (int s = 128; s >= 1; s >>= 1) {
    if (tid < s)
      for (int q = 0; q < 6; ++q) red[q][tid] += red[q][tid + s];
    __syncthreads();
  }

  if (tid == 0) {
    const bool valid = (nrm >= 1e-6f) && (cnt >= 2.5f);
    float A[3][3], Q[3][3];
    if (valid) {
      A[0][0] = red[0][0]; A[0][1] = red[1][0]; A[0][2] = red[2][0];
      A[1][0] = red[1][0]; A[1][1] = red[3][0]; A[1][2] = red[4][0];
      A[2][0] = red[2][0]; A[2][1] = red[4][0]; A[2][2] = red[5][0];
    } else {
      for (int i = 0; i < 3; ++i)
        for (int j = 0; j < 3; ++j) A[i][j] = (i == j) ? 1.f : 0.f;
    }
    for (int i = 0; i < 3; ++i)
      for (int j = 0; j < 3; ++j) Q[i][j] = (i == j) ? 1.f : 0.f;
    for (int sweep = 0; sweep < 10; ++sweep) {
      jacobi_rot(A, Q, 0, 1);
      jacobi_rot(A, Q, 0, 2);
      jacobi_rot(A, Q, 1, 2);
    }
    float w[3] = {A[0][0], A[1][1], A[2][2]};
    // Sort eigenpairs descending (columns of Q).
    for (int i = 0; i < 2; ++i)
      for (int j = 0; j < 2 - i; ++j)
        if (w[j] < w[j + 1]) {
          float tw = w[j]; w[j] = w[j + 1]; w[j + 1] = tw;
          for (int r = 0; r < 3; ++r) {
            float tq = Q[r][j]; Q[r][j] = Q[r][j + 1]; Q[r][j + 1] = tq;
          }
        }
    // Deterministic sign per eigenvector: largest-|.| component positive.
    for (int j = 0; j < 3; ++j) {
      int am = 0;
      if (fabsf(Q[1][j]) > fabsf(Q[am][j])) am = 1;
      if (fabsf(Q[2][j]) > fabsf(Q[am][j])) am = 2;
      if (Q[am][j] < 0.f)
        for (int r = 0; r < 3; ++r) Q[r][j] = -Q[r][j];
    }
    // rn = vh[:,2] = row 2 of V (z-components of the eigenvectors).
    float rnx = Q[2][0], rny = Q[2][1], rnz = Q[2][2];
    if (rnx * ux + rny * uy + rnz * uz < 0.f) { rnx = -rnx; rny = -rny; rnz = -rnz; }
    const float rd = -(mx * rnx + my * rny + mz * rnz);

    float* st = stats + ((size_t)b * M_ + m) * 12;
    st[0] = ux; st[1] = uy; st[2] = uz; st[3] = pd;
    st[4] = rnx; st[5] = rny; st[6] = rnz; st[7] = rd;
    st[8] = valid ? 1.f : 0.f;
  }
}

// ---------------------------------------------------------------------------
// 5) Finalize: per point, apply the 16 plane projections sequentially (mask
// from fixed pts_res), write refined + disp; block 0 copies planes through.
// ---------------------------------------------------------------------------
__global__ __launch_bounds__(256) void finalize_kernel(
    const float* __restrict__ pts_res, const float* __restrict__ stats,
    const float* __restrict__ planes,
    float* __restrict__ refined, float* __restrict__ disp,
    float* __restrict__ planes_out, int N_, int M_, int nPlaneElems) {
  const int gid = blockIdx.x * 256 + threadIdx.x;    // over B*N
  const int b = gid / N_;
  const float px = pts_res[(size_t)gid * 3];
  const float py = pts_res[(size_t)gid * 3 + 1];
  const float pz = pts_res[(size_t)gid * 3 + 2];
  float qx = px, qy = py, qz = pz;
  for (int m = 0; m < M_; ++m) {
    const float* s = stats + ((size_t)b * M_ + m) * 12;
    const float dist = fabsf(px * s[0] + py * s[1] + pz * s[2] + s[3]);
    if (dist < THRESH && s[8] > 0.5f) {
      const float dp = qx * s[4] + qy * s[5] + qz * s[6] + s[7];
      qx -= s[4] * dp; qy -= s[5] * dp; qz -= s[6] * dp;
    }
  }
  refined[(size_t)gid * 3]     = qx;
  refined[(size_t)gid * 3 + 1] = qy;
  refined[(size_t)gid * 3 + 2] = qz;
  disp[(size_t)gid * 3]     = qx - px;
  disp[(size_t)gid * 3 + 1] = qy - py;
  disp[(size_t)gid * 3 + 2] = qz - pz;
  if (blockIdx.x == 0 && threadIdx.x < nPlaneElems)
    planes_out[threadIdx.x] = planes[threadIdx.x];
}

// ---------------------------------------------------------------------------
extern "C" void kernel_launch(void* const* d_in, const int* in_sizes, int n_in,
                              void* d_out, int out_size, void* d_ws, size_t ws_size,
                              hipStream_t stream) {
  const int B = 2, N = 8192, M = 16;
  const size_t BN = (size_t)B * N;

  const float* points = (const float*)d_in[0];
  const float* planes = (const float*)d_in[1];
  const float* ec1_w1 = (const float*)d_in[2];
  const float* ec1_b1 = (const float*)d_in[3];
  const float* ec1_w2 = (const float*)d_in[4];
  const float* ec1_b2 = (const float*)d_in[5];
  const float* ec2_w1 = (const float*)d_in[6];
  const float* ec2_b1 = (const float*)d_in[7];
  const float* ec2_w2 = (const float*)d_in[8];
  const float* ec2_b2 = (const float*)d_in[9];
  const float* ec3_w1 = (const float*)d_in[10];
  const float* ec3_b1 = (const float*)d_in[11];
  const float* ec3_w2 = (const float*)d_in[12];
  const float* ec3_b2 = (const float*)d_in[13];
  const float* mlp_w1 = (const float*)d_in[14];
  const float* mlp_b1 = (const float*)d_in[15];
  const float* mlp_w2 = (const float*)d_in[16];
  const float* mlp_b2 = (const float*)d_in[17];

  // Workspace carve-up.
  int*   knn_idx = (int*)d_ws;                              // BN*16 ints
  float* f1      = (float*)(knn_idx + BN * KNN_K);          // BN*64
  float* f2      = f1 + BN * 64;
  float* f3      = f2 + BN * 64;
  float* pts_res = f3 + BN * 64;                            // BN*3
  float* stats   = pts_res + BN * 3;                        // B*M*12

  // Output carve-up: refined | residual | disp | planes.
  float* out        = (float*)d_out;
  float* o_refined  = out;
  float* o_residual = out + BN * 3;
  float* o_disp     = out + 2 * BN * 3;
  float* o_planes   = out + 3 * BN * 3;

  // 1) kNN.
  knn_kernel<<<dim3(N / 256, B), 256, 0, stream>>>(points, knn_idx, N);

  // 2) EdgeConv chain (same idx throughout, as in the reference).
  const int ecBlocks = (int)(BN / 4);
  edgeconv_kernel<3, 8><<<ecBlocks, 128, 0, stream>>>(
      points, knn_idx, ec1_w1, ec1_b1, ec1_w2, ec1_b2, f1, N);
  edgeconv_kernel<64, 128><<<ecBlocks, 128, 0, stream>>>(
      f1, knn_idx, ec2_w1, ec2_b1, ec2_w2, ec2_b2, f2, N);
  edgeconv_kernel<64, 128><<<ecBlocks, 128, 0, stream>>>(
      f2, knn_idx, ec3_w1, ec3_b1, ec3_w2, ec3_b2, f3, N);

  // 3) MLP head -> residual, pts_res.
  mlp_kernel<<<(int)(BN / 32), 64, 0, stream>>>(
      f1, f2, f3, mlp_w1, mlp_b1, mlp_w2, mlp_b2, points, o_residual, pts_res);

  // 4) Per-plane stats (+3x3 eigen).
  plane_stats_kernel<<<dim3(M, B), 256, 0, stream>>>(planes, pts_res, stats, N, M);

  // 5) Apply projections, write refined/disp/planes.
  finalize_kernel<<<(int)(BN / 256), 256, 0, stream>>>(
      pts_res, stats, planes, o_refined, o_disp, o_planes, N, M, B * M * 4);
}